// SARTlayer_18631568130335
// MI455X (gfx1250) — compile-verified
//
#include <hip/hip_runtime.h>
#include <math.h>

#define N_VIEWS 180
#define N_DET   512
#define IMG     512
#define N_S     768
#define EPS_F   1e-6f

typedef float v2f __attribute__((ext_vector_type(2)));
typedef float v8f __attribute__((ext_vector_type(8)));

__device__ __forceinline__ void view_trig(int view, float& c, float& si) {
    float theta = (float)((double)view * (M_PI / 180.0));
    __sincosf(theta, &si, &c);
}

__device__ __forceinline__ int clampi(int v, int lo, int hi) {
    return v < lo ? lo : (v > hi ? hi : v);
}

// ---------------------------------------------------------------------------
// Forward projection of one view + residual.
// Block = 256 threads (8 waves), covers 128 detector bins (16 per wave).
// Per wave: lane (half = lane>>4, row = lane&15) -> bin = base + row.
//   half 0 accumulates s-chunks 0,1 ; half 1 accumulates s-chunks 2,3
//   (192 samples each) -> exactly the ISA 32-bit 16x4 A-matrix layout:
//   lanes 0-15: VGPR0=K0, VGPR1=K1 ; lanes 16-31: VGPR0=K2, VGPR1=K3.
// One V_WMMA_F32_16X16X4_F32 with B = all-ones reduces K; since every B
// column is identical, every lane's 8 accumulator VGPRs hold the 8 row sums
// (lanes 0-15 -> bins base+0..7, lanes 16-31 -> bins base+8..15).
//
// Inner loop is fully branchless: tap addresses are clamped (always legal)
// and the out-of-range masks are folded into the bilinear weights, so the
// 4 taps issue as an unconditional load clause instead of 4 exec-mask
// branches per sample.
// ---------------------------------------------------------------------------
__global__ __launch_bounds__(256) void sart_project(
        const float* __restrict__ x,     // [B, IMG, IMG] current image
        const float* __restrict__ proj,  // [B, 1, N_VIEWS, N_DET]
        float* __restrict__ r,           // [B, N_DET] residual out
        int view)
{
    const int b    = blockIdx.y;
    const int lane = threadIdx.x & 31;
    const int wave = threadIdx.x >> 5;
    const int half = lane >> 4;
    const int row  = lane & 15;
    const int binbase = blockIdx.x * 128 + wave * 16;
    const int bin     = binbase + row;

    float c, si;
    view_trig(view, c, si);

    const float cx   = (IMG - 1) * 0.5f;
    const float tval = (float)bin - (N_DET - 1) * 0.5f;
    const float* img = x + (size_t)b * IMG * IMG;

    v2f ax, ao;   // A-matrix fragments: image-projection / ones-projection
    #pragma unroll
    for (int cidx = 0; cidx < 2; ++cidx) {
        const int chunk = 2 * half + cidx;         // this lane's K index
        float sval = (float)(chunk * 192) - (N_S - 1) * 0.5f;
        float X = tval * c  - sval * si + cx;
        float Y = tval * si + sval * c  + cx;
        float accx = 0.f, acco = 0.f;
        for (int s = 0; s < 192; ++s) {
            float fx = floorf(X), fy = floorf(Y);
            int ix0 = (int)fx, iy0 = (int)fy;
            float wx = X - fx,  wy = Y - fy;
            // masks folded into weights (branchless)
            float wx0 = ((unsigned)ix0       < (unsigned)IMG) ? (1.f - wx) : 0.f;
            float wx1 = ((unsigned)(ix0 + 1) < (unsigned)IMG) ? wx         : 0.f;
            float wy0 = ((unsigned)iy0       < (unsigned)IMG) ? (1.f - wy) : 0.f;
            float wy1 = ((unsigned)(iy0 + 1) < (unsigned)IMG) ? wy         : 0.f;
            // clamped addresses: always legal; masked weight kills garbage
            int x0c = clampi(ix0,     0, IMG - 1);
            int x1c = clampi(ix0 + 1, 0, IMG - 1);
            int y0c = clampi(iy0,     0, IMG - 1);
            int y1c = clampi(iy0 + 1, 0, IMG - 1);
            const float* r0 = img + y0c * IMG;
            const float* r1 = img + y1c * IMG;
            float v00 = r0[x0c];
            float v10 = r0[x1c];
            float v01 = r1[x0c];
            float v11 = r1[x1c];
            accx += (v00 * wx0 + v10 * wx1) * wy0
                  + (v01 * wx0 + v11 * wx1) * wy1;
            acco += (wx0 + wx1) * (wy0 + wy1);   // ones-image sample
            X -= si; Y += c;                     // step along the ray
        }
        ax[cidx] = accx;
        ao[cidx] = acco;
    }

    // K-reduction on the matrix pipe: D = A * ones(4x16)
    v2f bones; bones[0] = 1.f; bones[1] = 1.f;
    v8f z = {};
    v8f sx = __builtin_amdgcn_wmma_f32_16x16x4_f32(
        false, ax, false, bones, (short)0, z, false, false);
    v8f so = __builtin_amdgcn_wmma_f32_16x16x4_f32(
        false, ao, false, bones, (short)0, z, false, false);

    // lanes 0 and 16 of each wave write 8 residuals each
    if (row == 0) {
        const int base = binbase + half * 8;
        const float* p = proj + ((size_t)b * N_VIEWS + view) * N_DET + base;
        float* ro = r + b * N_DET + base;
        #pragma unroll
        for (int k = 0; k < 8; ++k)
            ro[k] = (p[k] - sx[k]) / (so[k] + EPS_F);
    }
}

// ---------------------------------------------------------------------------
// Backprojection update (exact adjoint, gather form, atomic-free).
// Pixel's rotated coordinates (u,v) in (detector-index, sample-index) space;
// only lattice points within L-inf distance 1 in image space contribute,
// which lie within L2 distance sqrt(2) in (t,s) space -> 3x3 candidates.
// den recomputes col_sums[view] (backprojection of ones) on the fly.
// ---------------------------------------------------------------------------
__global__ __launch_bounds__(256) void sart_update(
        float* __restrict__ x,          // [B, IMG, IMG] image (in place)
        const float* __restrict__ r,    // [B, N_DET]
        int view)
{
    __shared__ float rsh[N_DET];
    const int b = blockIdx.z;
    for (int i = threadIdx.x; i < N_DET; i += 256)
        rsh[i] = r[b * N_DET + i];
    __syncthreads();

    const int tx = threadIdx.x & 15, ty = threadIdx.x >> 4;
    const int ix = blockIdx.x * 16 + tx;
    const int iy = blockIdx.y * 16 + ty;

    float c, si;
    view_trig(view, c, si);

    const float dx = (float)ix - (IMG - 1) * 0.5f;
    const float dy = (float)iy - (IMG - 1) * 0.5f;
    const float u = dx * c + dy * si + (N_DET - 1) * 0.5f;  // detector coord
    const float v = -dx * si + dy * c + (N_S - 1) * 0.5f;   // sample coord

    const int k0 = (int)ceilf(u - 1.4143f);
    const int m0 = (int)ceilf(v - 1.4143f);

    float num = 0.f, den = 0.f;
    #pragma unroll
    for (int dk = 0; dk < 3; ++dk) {
        const int k = k0 + dk;
        const bool vk = (unsigned)k < (unsigned)N_DET;
        const int kc = clampi(k, 0, N_DET - 1);
        const float rv = rsh[kc];
        const float dt = (float)k - u;          // t-offset (rotated frame)
        #pragma unroll
        for (int dm = 0; dm < 3; ++dm) {
            const int m = m0 + dm;
            const float ds = (float)m - v;      // s-offset (rotated frame)
            const float ex = dt * c - ds * si;  // image-space x displacement
            const float ey = dt * si + ds * c;  // image-space y displacement
            float w = fmaxf(0.f, 1.f - fabsf(ex)) * fmaxf(0.f, 1.f - fabsf(ey));
            const bool vm = (unsigned)m < (unsigned)N_S;
            w = (vk && vm) ? w : 0.f;
            num += w * rv;
            den += w;
        }
    }

    const size_t idx = (size_t)b * IMG * IMG + (size_t)iy * IMG + ix;
    x[idx] += num / (den + EPS_F);              // LAM = 1
}

// ---------------------------------------------------------------------------
extern "C" void kernel_launch(void* const* d_in, const int* in_sizes, int n_in,
                              void* d_out, int out_size, void* d_ws, size_t ws_size,
                              hipStream_t stream)
{
    const float* input = (const float*)d_in[0];   // [2,1,512,512]
    const float* proj  = (const float*)d_in[1];   // [2,1,180,512]
    float* img  = (float*)d_out;                  // iterate in place
    float* r_ws = (float*)d_ws;                   // [2,512] residual

    hipMemcpyAsync(img, input, (size_t)2 * IMG * IMG * sizeof(float),
                   hipMemcpyDeviceToDevice, stream);

    for (int vw = 0; vw < N_VIEWS; ++vw) {
        sart_project<<<dim3(4, 2), 256, 0, stream>>>(img, proj, r_ws, vw);
        sart_update<<<dim3(32, 32, 2), 256, 0, stream>>>(img, r_ws, vw);
    }
}